// ExplicitRegisters_70231305224215
// MI455X (gfx1250) — compile-verified
//
#include <hip/hip_runtime.h>
#include <math.h>

#define B_ 8
#define L_ 2048
#define D_ 1024
#define CH 32
#define NC (L_/CH)          // 64 chunks per sequence

typedef __attribute__((ext_vector_type(2))) float v2f;
typedef __attribute__((ext_vector_type(8))) float v8f;

// ---- LDS float layout for k1 (all offsets in floats) ----
// WB[col][d] : 64 x 1024 weights in B-operand layout, pitch 1041
//   bank = (17*col + d) % 64 -> conflict-free for both B-fragment half-waves.
#define WB_      0
#define WB_PITCH 1041
#define WB_SZ    (64*WB_PITCH)        // 66624
#define GOFF     (WB_ + WB_SZ)        // ln_g [1024]
#define BOFF     (GOFF + 1024)        // ln_b [1024]
#define HBUF     (BOFF + 1024)        // per-wave 1088: hn[64][17] / logits[16][68]
#define HPITCH   17
#define LPITCH   68
#define STATS    (HBUF + 8*1088)      // per-wave 32: (mu,rinv) x 16 tokens
#define SM1_FLOATS (STATS + 8*32)     // 77632 floats = 310528 B  (< 320KB/WGP)

__device__ __forceinline__ float wave_sum32(float v) {
  v += __shfl_xor(v, 16, 32);
  v += __shfl_xor(v, 8, 32);
  v += __shfl_xor(v, 4, 32);
  v += __shfl_xor(v, 2, 32);
  v += __shfl_xor(v, 1, 32);
  return v;
}

// -------- Kernel 1: LN + all 4 projections as one WMMA GEMM + activations --------
// grid 128 x 256; wave owns 16 tokens. Logit cols: [0,8)=rq, [8,16)=wq, [16,48)=wv,
// 48=wg, [49,64)=zero pad. C = hn[16x1024] @ W[1024x64] via V_WMMA_F32_16X16X4_F32.
__global__ __launch_bounds__(256) void k1_proj(
    const float* __restrict__ x, const float* __restrict__ ln_g, const float* __restrict__ ln_b,
    const float* __restrict__ rq_w, const float* __restrict__ rq_b,
    const float* __restrict__ wq_w, const float* __restrict__ wq_b,
    const float* __restrict__ wv_w, const float* __restrict__ wv_b,
    const float* __restrict__ wg_w, const float* __restrict__ wg_b,
    float* __restrict__ rw_o, float* __restrict__ us_o, float* __restrict__ wv_o)
{
  extern __shared__ float sm[];
  const int tid = threadIdx.x;

  // Stage weights once per block into B-operand layout (transposed to [col][d]).
  for (int i = tid; i < 64*1024; i += 256) {
    const int c = i >> 10, d = i & 1023;
    float w;
    if (c < 8)        w = rq_w[d*8 + c];
    else if (c < 16)  w = wq_w[d*8 + (c-8)];
    else if (c < 48)  w = wv_w[d*32 + (c-16)];
    else if (c == 48) w = wg_w[d];
    else              w = 0.f;
    sm[WB_ + c*WB_PITCH + d] = w;
  }
  for (int d = tid; d < 1024; d += 256) {
    sm[GOFF + d] = ln_g[d];
    sm[BOFF + d] = ln_b[d];
  }
  __syncthreads();

  const int wave = tid >> 5, lane = tid & 31;
  const int ml = lane & 15;      // M (A) / N (B) lane index
  const int hi = lane >> 4;      // half-wave: K pair {0,1} vs {2,3}
  const int kk = hi * 2;
  float* hb    = &sm[HBUF + wave*1088];
  float* stats = &sm[STATS + wave*32];
  const int t0 = blockIdx.x*128 + wave*16;

  // ---- LN statistics for the wave's 16 tokens ----
  for (int tt = 0; tt < 16; ++tt) {
    float s = 0.f, q = 0.f;
    const size_t xb = (size_t)(t0 + tt) * D_;
    for (int j = 0; j < 32; ++j) {
      const float v = x[xb + j*32 + lane];
      s += v; q += v*v;
    }
    s = wave_sum32(s); q = wave_sum32(q);
    const float mu = s * (1.0f/1024.0f);
    const float ri = rsqrtf(q*(1.0f/1024.0f) - mu*mu + 1e-5f);
    if (lane == 0) { stats[tt*2] = mu; stats[tt*2+1] = ri; }
  }
  __builtin_amdgcn_wave_barrier();

  v8f C[4];
  #pragma unroll
  for (int ct = 0; ct < 4; ++ct)
    #pragma unroll
    for (int i = 0; i < 8; ++i) C[ct][i] = 0.f;

  // ---- stream d in blocks of 64: build hn tile, then 16 WMMA k-steps ----
  for (int dblk = 0; dblk < 16; ++dblk) {
    const int dg0 = dblk * 64;
    // fill hn[64][17]: writes at (jj*32+lane)*17+tt -> 17 invertible mod 64, conflict-free
    #pragma unroll
    for (int jj = 0; jj < 2; ++jj) {
      const int dl = jj*32 + lane;
      const float gv = sm[GOFF + dg0 + dl];
      const float bv = sm[BOFF + dg0 + dl];
      for (int tt = 0; tt < 16; ++tt) {
        const float mu = stats[tt*2], ri = stats[tt*2+1];
        const float v  = x[(size_t)(t0 + tt)*D_ + dg0 + dl];   // 128B coalesced (L2 hit)
        hb[dl*HPITCH + tt] = (v - mu) * ri * gv + bv;
      }
    }
    __builtin_amdgcn_wave_barrier();

    for (int s = 0; s < 16; ++s) {
      const int dls = s*4 + kk;
      v2f a;
      a.x = hb[dls*HPITCH + ml];            // A: 16x4 tile of hn
      a.y = hb[(dls+1)*HPITCH + ml];
      const int dg = dg0 + dls;
      #pragma unroll
      for (int ct = 0; ct < 4; ++ct) {
        v2f b;                               // B: 4x16 tile of W (conflict-free pitch)
        b.x = sm[WB_ + (ct*16 + ml)*WB_PITCH + dg];
        b.y = sm[WB_ + (ct*16 + ml)*WB_PITCH + dg + 1];
        C[ct] = __builtin_amdgcn_wmma_f32_16x16x4_f32(
                    false, a, false, b, (short)0, C[ct], false, false);
      }
    }
    __builtin_amdgcn_wave_barrier();   // DS in-order per wave: next fill can't pass A-reads
  }

  // ---- scatter logits to LDS (reuse hb as [16][68]) ----
  #pragma unroll
  for (int ct = 0; ct < 4; ++ct)
    #pragma unroll
    for (int i = 0; i < 8; ++i)
      hb[(i + 8*hi)*LPITCH + ct*16 + ml] = C[ct][i];   // D layout: VGPR i -> M=i+8*hi
  __builtin_amdgcn_wave_barrier();

  // ---- activations + coalesced stores, one token per iteration ----
  for (int tt = 0; tt < 16; ++tt) {
    const int t = t0 + tt;
    const float* lg = &hb[tt*LPITCH];
    float m1 = -1e30f, m2 = -1e30f;
    float l1[8], l2[8];
    #pragma unroll
    for (int k = 0; k < 8; ++k) {
      l1[k] = lg[k]     + rq_b[k];
      l2[k] = lg[8 + k] + wq_b[k];
      m1 = fmaxf(m1, l1[k]); m2 = fmaxf(m2, l2[k]);
    }
    float z1 = 0.f, z2 = 0.f;
    #pragma unroll
    for (int k = 0; k < 8; ++k) { z1 += expf(l1[k]-m1); z2 += expf(l2[k]-m2); }
    const float gate = 1.0f / (1.0f + expf(-(lg[48] + wg_b[0])));

    if (lane < 8)
      rw_o[(size_t)t*8 + lane] = expf(lg[lane] + rq_b[lane] - m1) / z1;
    else if (lane < 16) {
      const int k = lane - 8;
      us_o[(size_t)t*8 + k] = expf(lg[8+k] + wq_b[k] - m2) / z2 * gate;
    }
    wv_o[(size_t)t*32 + lane] = lg[16 + lane] + wv_b[lane];
    __builtin_amdgcn_wave_barrier();
  }
}

// -------- Kernel 2a: per-chunk local recurrence summaries --------
// wave = (batch, chunk); lane = r. st <- (1-u)st + u*wv from 0; pa = prod(1-u).
__global__ __launch_bounds__(256) void k2a_local(
    const float* __restrict__ us, const float* __restrict__ wv,
    float* __restrict__ paA, float* __restrict__ Bc)
{
  extern __shared__ float sm[];
  const int wave = threadIdx.x >> 5, lane = threadIdx.x & 31;
  const int gw = blockIdx.x * 8 + wave;       // 0..511
  const int b  = gw >> 6, ch = gw & 63;
  const size_t tf0 = (size_t)b * L_ + (size_t)ch * CH;

  float* lus = &sm[wave * 256];
  for (int i = lane; i < 256; i += 32) lus[i] = us[tf0*8 + i];
  __builtin_amdgcn_wave_barrier();

  float st[8], pa[8];
  #pragma unroll
  for (int k = 0; k < 8; ++k) { st[k] = 0.f; pa[k] = 1.f; }
  for (int s = 0; s < CH; ++s) {
    const float wvv = wv[(tf0 + s)*32 + lane];
    #pragma unroll
    for (int k = 0; k < 8; ++k) {
      const float u = lus[s*8 + k];
      st[k] += u * (wvv - st[k]);
      pa[k] *= (1.0f - u);
    }
  }
  #pragma unroll
  for (int k = 0; k < 8; ++k) Bc[((size_t)gw*8 + k)*32 + lane] = st[k];
  if (lane == 0) {
    #pragma unroll
    for (int k = 0; k < 8; ++k) paA[gw*8 + k] = pa[k];
  }
}

// -------- Kernel 2b: sequential chunk prefix (tiny). wave = batch, lane = r --------
__global__ __launch_bounds__(256) void k2b_prefix(
    const float* __restrict__ paA, const float* __restrict__ Bc,
    float* __restrict__ Ss)
{
  const int wave = threadIdx.x >> 5, lane = threadIdx.x & 31;
  const int b = wave;
  float S[8];
  #pragma unroll
  for (int k = 0; k < 8; ++k) S[k] = 0.f;
  for (int c = 0; c < NC; ++c) {
    const int gw = b*NC + c;
    #pragma unroll
    for (int k = 0; k < 8; ++k) {
      Ss[((size_t)gw*8 + k)*32 + lane] = S[k];
      S[k] = paA[gw*8 + k] * S[k] + Bc[((size_t)gw*8 + k)*32 + lane];
    }
  }
}

// -------- Kernel 2c: replay chunks from start-state, emit rv (read-before-write) --------
__global__ __launch_bounds__(256) void k2c_emit(
    const float* __restrict__ rw, const float* __restrict__ us,
    const float* __restrict__ wv, const float* __restrict__ Ss,
    float* __restrict__ rv)
{
  extern __shared__ float sm[];
  const int wave = threadIdx.x >> 5, lane = threadIdx.x & 31;
  const int gw = blockIdx.x * 8 + wave;
  const int b  = gw >> 6, ch = gw & 63;
  const size_t tf0 = (size_t)b * L_ + (size_t)ch * CH;

  float* lrw = &sm[wave * 512];
  float* lus = lrw + 256;
  for (int i = lane; i < 256; i += 32) {
    lrw[i] = rw[tf0*8 + i];
    lus[i] = us[tf0*8 + i];
  }
  __builtin_amdgcn_wave_barrier();

  float st[8];
  #pragma unroll
  for (int k = 0; k < 8; ++k) st[k] = Ss[((size_t)gw*8 + k)*32 + lane];

  for (int s = 0; s < CH; ++s) {
    float acc = 0.f;
    #pragma unroll
    for (int k = 0; k < 8; ++k) acc += lrw[s*8 + k] * st[k];     // read BEFORE write
    rv[(tf0 + s)*32 + lane] = acc;
    const float wvv = wv[(tf0 + s)*32 + lane];
    #pragma unroll
    for (int k = 0; k < 8; ++k) {
      const float u = lus[s*8 + k];
      st[k] += u * (wvv - st[k]);
    }
  }
}

// -------- Kernel 3: out = mix * rv @ rp_w + rp_b via V_WMMA_F32_16X16X4_F32 --------
__global__ __launch_bounds__(256) void k3_wmma_out(
    const float* __restrict__ rv, const float* __restrict__ rp_w,
    const float* __restrict__ rp_b, const float* __restrict__ mixp,
    float* __restrict__ out)
{
  const int wave = threadIdx.x >> 5, lane = threadIdx.x & 31;
  const int m0 = blockIdx.y * 16;
  const int n0 = blockIdx.x * 128 + wave * 16;
  const int ml = lane & 15;
  const int hi = lane >> 4;
  const int kk = hi * 2;

  v2f a[8];
  #pragma unroll
  for (int s = 0; s < 8; ++s) {
    const float* p = &rv[(size_t)(m0 + ml)*32 + s*4 + kk];
    a[s].x = p[0];
    a[s].y = p[1];
  }

  v8f c = {0.f,0.f,0.f,0.f,0.f,0.f,0.f,0.f};
  #pragma unroll
  for (int s = 0; s < 8; ++s) {
    v2f bv;
    const int kb = s*4 + kk;
    bv.x = rp_w[(size_t)kb*1024 + n0 + ml];
    bv.y = rp_w[(size_t)(kb+1)*1024 + n0 + ml];
    c = __builtin_amdgcn_wmma_f32_16x16x4_f32(
            false, a[s], false, bv, (short)0, c, false, false);
  }

  const float mixv = mixp[0];
  const float bias = rp_b[n0 + ml];
  #pragma unroll
  for (int i = 0; i < 8; ++i) {
    out[(size_t)(m0 + i + hi*8)*1024 + n0 + ml] = mixv * c[i] + bias;
  }
}

extern "C" void kernel_launch(void* const* d_in, const int* in_sizes, int n_in,
                              void* d_out, int out_size, void* d_ws, size_t ws_size,
                              hipStream_t stream) {
  const float* x    = (const float*)d_in[0];
  const float* ln_g = (const float*)d_in[1];
  const float* ln_b = (const float*)d_in[2];
  const float* rq_w = (const float*)d_in[3];
  const float* rq_b = (const float*)d_in[4];
  const float* rp_w = (const float*)d_in[5];
  const float* rp_b = (const float*)d_in[6];
  const float* wq_w = (const float*)d_in[7];
  const float* wq_b = (const float*)d_in[8];
  const float* wv_w = (const float*)d_in[9];
  const float* wv_b = (const float*)d_in[10];
  const float* wg_w = (const float*)d_in[11];
  const float* wg_b = (const float*)d_in[12];
  const float* mix  = (const float*)d_in[13];

  float* ws  = (float*)d_ws;
  float* rw  = ws;                 // [BL,8]      131072
  float* us  = rw  + 131072;       // [BL,8]      131072
  float* wv  = us  + 131072;       // [BL,32]     524288
  float* rv  = wv  + 524288;       // [BL,32]     524288
  float* paA = rv  + 524288;       // [B*NC,8]    4096
  float* Bc  = paA + 4096;         // [B*NC,8,32] 131072
  float* Ss  = Bc  + 131072;       // [B*NC,8,32] 131072
  float* out = (float*)d_out;

  const size_t sm1 = (size_t)SM1_FLOATS * sizeof(float);  // ~303 KB dynamic LDS
  k1_proj<<<128, 256, sm1, stream>>>(x, ln_g, ln_b, rq_w, rq_b, wq_w, wq_b,
                                     wv_w, wv_b, wg_w, wg_b, rw, us, wv);
  k2a_local<<<64, 256, 8*256*sizeof(float), stream>>>(us, wv, paA, Bc);
  k2b_prefix<<<1, 256, 0, stream>>>(paA, Bc, Ss);
  k2c_emit<<<64, 256, 8*512*sizeof(float), stream>>>(rw, us, wv, Ss, rv);
  k3_wmma_out<<<dim3(8, 1024), 256, 0, stream>>>(rv, rp_w, rp_b, mix, out);
}